// SparseGraphConv_13262859010733
// MI455X (gfx1250) — compile-verified
//
#include <hip/hip_runtime.h>

typedef __attribute__((ext_vector_type(2))) float v2f;
typedef __attribute__((ext_vector_type(8))) float v8f;

#define CIN   64
#define COUT  64
#define TSTEP 16
#define NBATCH 2
#define FEAT  (TSTEP * COUT)          // 1024 floats per (node, batch)

// ---------------------------------------------------------------------------
// Kernel 1: zero the output (segment_sum identity; harness poisons d_out).
// ---------------------------------------------------------------------------
__global__ __launch_bounds__(256) void zero_f4(float4* __restrict__ p, int n4) {
    int i = blockIdx.x * blockDim.x + threadIdx.x;
    int stride = gridDim.x * blockDim.x;
    float4 z = make_float4(0.f, 0.f, 0.f, 0.f);
    for (; i < n4; i += stride) p[i] = z;
}

// ---------------------------------------------------------------------------
// Kernel 2: y[m][o] = sum_c x[m][c] * W[o][c] + b[o], m = 0..Mrows-1
// One wave -> 16 rows x 64 cols using V_WMMA_F32_16X16X4_F32.
//   A (16x4 f32): lanes 0-15 rows m0..m0+15, vgpr{0,1} = K {0,1}(half0)/{2,3}(half1)
//   B (4x16 f32): lane n holds W[o0+n][k0+2*half + {0,1}]
//   D (16x16 f32): vgpr r -> row m0+r (lanes 0-15) / m0+8+r (lanes 16-31), N=lane%16
// ---------------------------------------------------------------------------
__global__ __launch_bounds__(256) void linear_wmma(const float* __restrict__ x,
                                                   const float* __restrict__ W,
                                                   const float* __restrict__ bias,
                                                   float* __restrict__ y) {
    const int wave = threadIdx.x >> 5;
    const int lane = threadIdx.x & 31;
    const int half = lane >> 4;        // 0: K pair {0,1}; 1: K pair {2,3}
    const int l16  = lane & 15;

    const int rowTile = blockIdx.x * 8 + wave;   // 8 waves per block
    const int m0 = rowTile << 4;                 // 16 rows per wave

    v8f acc[4];
#pragma unroll
    for (int t = 0; t < 4; ++t) {
        float bv = bias[t * 16 + l16];
        acc[t] = (v8f){bv, bv, bv, bv, bv, bv, bv, bv};
    }

    const float* xrow = x + (size_t)(m0 + l16) * CIN + half * 2;
#pragma unroll
    for (int k0 = 0; k0 < CIN; k0 += 4) {
        v2f a = *(const v2f*)(xrow + k0);
#pragma unroll
        for (int t = 0; t < 4; ++t) {
            v2f bm = *(const v2f*)(W + (size_t)(t * 16 + l16) * CIN + k0 + half * 2);
            acc[t] = __builtin_amdgcn_wmma_f32_16x16x4_f32(
                false, a, false, bm, (short)0, acc[t], false, false);
        }
    }

#pragma unroll
    for (int t = 0; t < 4; ++t) {
        float* yout = y + (size_t)(m0 + half * 8) * COUT + t * 16 + l16;
#pragma unroll
        for (int r = 0; r < 8; ++r)
            yout[(size_t)r * COUT] = acc[t][r];
    }
}

// ---------------------------------------------------------------------------
// Kernel 3: for edge e: out[b, row, t, c] += val * y[b, col, t, c]
// One block per edge; 256 threads stride the 2*1024 feature elements.
// y and out both fit in the 192MB L2 -> gathers + f32 atomics stay on-chip.
// ---------------------------------------------------------------------------
__global__ __launch_bounds__(256) void spmm_scatter(const float* __restrict__ y,
                                                    const float* __restrict__ eval,
                                                    const int*   __restrict__ erow,
                                                    const int*   __restrict__ ecol,
                                                    float* __restrict__ out,
                                                    int nNodes) {
    const int e = blockIdx.x;
    const int r = erow[e];
    const int c = ecol[e];
    const float v = eval[e];

    const size_t bstride = (size_t)nNodes * FEAT;
    const float* src = y   + (size_t)c * FEAT;
    float*       dst = out + (size_t)r * FEAT;

#pragma unroll
    for (int i = 0; i < NBATCH * FEAT; i += 256) {
        int idx = i + threadIdx.x;
        int b = idx >> 10;              // batch
        int f = idx & (FEAT - 1);       // t*64 + c_out
        atomicAdd(dst + (size_t)b * bstride + f, v * src[(size_t)b * bstride + f]);
    }
}

// ---------------------------------------------------------------------------
// d_in order (setup_inputs dict): x, W, b, edge_val, edge_row, edge_col
// ---------------------------------------------------------------------------
extern "C" void kernel_launch(void* const* d_in, const int* in_sizes, int n_in,
                              void* d_out, int out_size, void* d_ws, size_t ws_size,
                              hipStream_t stream) {
    const float* x    = (const float*)d_in[0];
    const float* W    = (const float*)d_in[1];
    const float* bias = (const float*)d_in[2];
    const float* ev   = (const float*)d_in[3];
    const int*   er   = (const int*)d_in[4];
    const int*   ec   = (const int*)d_in[5];
    float* out = (float*)d_out;
    float* y   = (float*)d_ws;                 // 320000 x 64 f32 = 81.92 MB

    const int E      = in_sizes[3];            // 160000 edges
    const int Mrows  = in_sizes[0] / CIN;      // B*N*T = 320000
    const int nNodes = out_size / (NBATCH * FEAT);  // 10000

    // 1) out = 0
    int n4 = out_size / 4;
    zero_f4<<<2048, 256, 0, stream>>>((float4*)out, n4);

    // 2) y = x @ W^T + b   (WMMA f32)
    int gemmBlocks = Mrows / (16 * 8);         // 16 rows/wave, 8 waves/block
    linear_wmma<<<gemmBlocks, 256, 0, stream>>>(x, W, bias, y);

    // 3) scatter-add over edges
    spmm_scatter<<<E, 256, 0, stream>>>(y, ev, er, ec, out, nNodes);
}